// SubSamplingLayer_18494129177023
// MI455X (gfx1250) — compile-verified
//
#include <hip/hip_runtime.h>
#include <hip/hip_bf16.h>
#include <math.h>

// ---------------- problem constants ----------------
#define RESN   320
#define OSN    400            // ceil(1.25 * 320)
#define NG2    (OSN * OSN)    // 160000
#define BATCH  16
#define MPTS   25600          // RES*RES/4

__device__ __constant__ float kBeta  = 6.9966584f;   // pi*sqrt((4/1.25*0.75)^2 - 0.8)
#define TWO_PI_OVER_N 0.015707963267948967f          // 2*pi/400

// ---------------- WMMA types / wrapper ----------------
typedef float v2f __attribute__((ext_vector_type(2)));
typedef float v8f __attribute__((ext_vector_type(8)));

__device__ __forceinline__ v8f wmma_f32_4(v2f a, v2f b, v8f c) {
  // D(16x16,f32) = A(16x4,f32) * B(4x16,f32) + C
  return __builtin_amdgcn_wmma_f32_16x16x4_f32(
      /*neg_a=*/false, a, /*neg_b=*/false, b,
      /*c_mod=*/(short)0, c, /*reuse_a=*/false, /*reuse_b=*/false);
}

// ---------------- modified Bessel I0 (A&S 9.8.1 / 9.8.2) ----------------
__device__ __forceinline__ float i0f_dev(float x) {
  if (x < 3.75f) {
    float t = (x * x) * (1.0f / (3.75f * 3.75f));
    return 1.0f + t * (3.5156229f + t * (3.0899424f + t * (1.2067492f +
                 t * (0.2659732f + t * (0.0360768f + t * 0.0045813f)))));
  } else {
    float t = 3.75f / x;
    float p = 0.39894228f + t * (0.01328592f + t * (0.00225319f + t * (-0.00157565f +
              t * (0.00916281f + t * (-0.02057706f + t * (0.02635537f +
              t * (-0.01647633f + t * 0.00392377f)))))));
    return __expf(x) * p * rsqrtf(x);
  }
}

// ---------------- kernel 0: twiddle tables W[k][n] = exp(+2*pi*i*k*n/400) ----------------
__global__ __launch_bounds__(256) void k_twiddle(float* __restrict__ Wre,
                                                 float* __restrict__ Wim) {
  int tid = blockIdx.x * 256 + threadIdx.x;        // 625 blocks -> exactly 160000
  int k = tid / OSN;
  int n = tid - k * OSN;
  int r = (k * n) % OSN;                           // keep the angle small & exact
  float s, c;
  __sincosf((float)r * TWO_PI_OVER_N, &s, &c);
  Wre[tid] = c;
  Wim[tid] = s;
}

// ---------------- kernel 1: bilinear sample + Kaiser-Bessel scatter (ifftshift folded in) ----
__global__ __launch_bounds__(256) void k_sample_scatter(const float* __restrict__ ksp,
                                                        const float* __restrict__ traj,
                                                        float* __restrict__ grid) {
  int tid = blockIdx.x * 256 + threadIdx.x;        // 1600 blocks -> exactly 409600
  int b = tid & (BATCH - 1);
  int m = tid >> 4;

  float t0 = traj[2 * m + 0];                      // dim-0 coord for NUFFT, gx for sampling
  float t1 = traj[2 * m + 1];

  // ---- bilinear grid_sample (align-corners style mapping from reference) ----
  float px = (t0 * (1.0f / 160.0f) + 1.0f) * 0.5f * 319.0f;
  float py = (t1 * (1.0f / 160.0f) + 1.0f) * 0.5f * 319.0f;
  float x0 = floorf(px), y0 = floorf(py);
  float fx = px - x0,    fy = py - y0;

  float yre = 0.0f, yim = 0.0f;
  const float* kb = ksp + (size_t)b * RESN * RESN * 2;
#pragma unroll
  for (int dy = 0; dy < 2; ++dy) {
#pragma unroll
    for (int dx = 0; dx < 2; ++dx) {
      float xi = x0 + (float)dx;
      float yi = y0 + (float)dy;
      float w = (dx ? fx : 1.0f - fx) * (dy ? fy : 1.0f - fy);
      if (xi >= 0.0f && xi < (float)RESN && yi >= 0.0f && yi < (float)RESN) {
        int base = (((int)yi) * RESN + (int)xi) * 2;
        yre += kb[base + 0] * w;
        yim += kb[base + 1] * w;
      }
    }
  }

  // ---- Kaiser-Bessel gridding weights (WIDTH=4 -> 4x4 footprint) ----
  float c0 = t0 * 1.25f + 200.0f;
  float c1 = t1 * 1.25f + 200.0f;
  float s0 = ceilf(c0 - 2.0f);
  float s1 = ceilf(c1 - 2.0f);
  int s0i = (int)s0, s1i = (int)s1;

  float w0[4], w1[4];
  int   r0[4], r1[4];
#pragma unroll
  for (int j = 0; j < 4; ++j) {
    float d0 = (c0 - (s0 + (float)j)) * 0.5f;
    float d1 = (c1 - (s1 + (float)j)) * 0.5f;
    float q0 = fmaxf(1.0f - d0 * d0, 0.0f);
    float q1 = fmaxf(1.0f - d1 * d1, 0.0f);
    w0[j] = i0f_dev(kBeta * sqrtf(q0)) * 0.25f;
    w1[j] = i0f_dev(kBeta * sqrtf(q1)) * 0.25f;
    r0[j] = (s0i + j + 200) % OSN;                 // ifftshift folded into index
    r1[j] = (s1i + j + 200) % OSN;
  }

  float* gre = grid + (size_t)b * 2 * NG2;
  float* gim = gre + NG2;
#pragma unroll
  for (int j0 = 0; j0 < 4; ++j0) {
#pragma unroll
    for (int j1 = 0; j1 < 4; ++j1) {
      float w = w0[j0] * w1[j1];
      int o = r0[j0] * OSN + r1[j1];
      atomicAdd(gre + o, yre * w);
      atomicAdd(gim + o, yim * w);
    }
  }
}

// ---------------- kernel 2: stage 1, T = G * W (complex, row transform) ------------------
// one wave per 16x16 output tile; 10000 tiles = 1250 blocks * 8 waves
__global__ __launch_bounds__(256) void k_dft_stage1(const float* __restrict__ grid,
                                                    const float* __restrict__ Wre,
                                                    const float* __restrict__ Wim,
                                                    float* __restrict__ T) {
  int wave = threadIdx.x >> 5;
  int tile = blockIdx.x * 8 + wave;
  int b  = tile / 625;
  int rm = tile - b * 625;
  int tm = rm / 25;
  int tn = rm - tm * 25;

  int lane = threadIdx.x & 31;
  int half = lane >> 4;
  int l16  = lane & 15;
  int row  = tm * 16 + l16;        // A-fragment row
  int coln = tn * 16 + l16;        // B-fragment / output column

  const float* Gre = grid + (size_t)b * 2 * NG2;
  const float* Gim = Gre + NG2;

  v8f accRe = {};
  v8f accIm = {};

  for (int k0 = 0; k0 < OSN; k0 += 4) {
    int kk = k0 + 2 * half;
    v2f a_re = *(const v2f*)(Gre + row * OSN + kk);
    v2f a_im = *(const v2f*)(Gim + row * OSN + kk);
    v2f b_re, b_im;
    b_re.x = Wre[kk * OSN + coln];
    b_re.y = Wre[(kk + 1) * OSN + coln];
    b_im.x = Wim[kk * OSN + coln];
    b_im.y = Wim[(kk + 1) * OSN + coln];
    v2f b_imn = -b_im;             // f32 WMMA has no A/B neg -> VALU negate
    accRe = wmma_f32_4(a_re, b_re,  accRe);
    accRe = wmma_f32_4(a_im, b_imn, accRe);
    accIm = wmma_f32_4(a_re, b_im,  accIm);
    accIm = wmma_f32_4(a_im, b_re,  accIm);
  }

  float* Tre = T + (size_t)b * 2 * NG2;
  float* Tim = Tre + NG2;
#pragma unroll
  for (int v = 0; v < 8; ++v) {
    int orow = tm * 16 + v + 8 * half;
    Tre[orow * OSN + coln] = accRe[v];
    Tim[orow * OSN + coln] = accIm[v];
  }
}

// ---------------- kernel 3: stage 2, I = W * T (complex, column transform) ---------------
__global__ __launch_bounds__(256) void k_dft_stage2(const float* __restrict__ T,
                                                    const float* __restrict__ Wre,
                                                    const float* __restrict__ Wim,
                                                    float* __restrict__ I) {
  int wave = threadIdx.x >> 5;
  int tile = blockIdx.x * 8 + wave;
  int b  = tile / 625;
  int rm = tile - b * 625;
  int tm = rm / 25;
  int tn = rm - tm * 25;

  int lane = threadIdx.x & 31;
  int half = lane >> 4;
  int l16  = lane & 15;
  int kp   = tm * 16 + l16;        // output row k' (A row of W)
  int coln = tn * 16 + l16;        // output column

  const float* Tre = T + (size_t)b * 2 * NG2;
  const float* Tim = Tre + NG2;

  v8f accRe = {};
  v8f accIm = {};

  for (int k0 = 0; k0 < OSN; k0 += 4) {
    int kk = k0 + 2 * half;
    v2f a_wre = *(const v2f*)(Wre + kp * OSN + kk);
    v2f a_wim = *(const v2f*)(Wim + kp * OSN + kk);
    v2f a_wimn = -a_wim;
    v2f b_tre, b_tim;
    b_tre.x = Tre[kk * OSN + coln];
    b_tre.y = Tre[(kk + 1) * OSN + coln];
    b_tim.x = Tim[kk * OSN + coln];
    b_tim.y = Tim[(kk + 1) * OSN + coln];
    // I_re = Wre*Tre - Wim*Tim ; I_im = Wre*Tim + Wim*Tre
    accRe = wmma_f32_4(a_wre,  b_tre, accRe);
    accRe = wmma_f32_4(a_wimn, b_tim, accRe);
    accIm = wmma_f32_4(a_wre,  b_tim, accIm);
    accIm = wmma_f32_4(a_wim,  b_tre, accIm);
  }

  float* Ire = I + (size_t)b * 2 * NG2;
  float* Iim = Ire + NG2;
#pragma unroll
  for (int v = 0; v < 8; ++v) {
    int orow = tm * 16 + v + 8 * half;
    Ire[orow * OSN + coln] = accRe[v];
    Iim[orow * OSN + coln] = accIm[v];
  }
}

// ---------------- kernel 4: fftshift + crop + scale + apodization ------------------------
__global__ __launch_bounds__(256) void k_finalize(const float* __restrict__ I,
                                                  float* __restrict__ out) {
  int tid = blockIdx.x * 256 + threadIdx.x;        // 6400 blocks -> exactly 16*320*320
  int x = tid % RESN;
  int y = (tid / RESN) % RESN;
  int b = tid / (RESN * RESN);

  int row = (240 + y) % OSN;                       // fftshift(+200) then crop offset(+40)
  int col = (240 + x) % OSN;

  const float* Ire = I + (size_t)b * 2 * NG2;
  const float* Iim = Ire + NG2;
  float re = Ire[row * OSN + col];
  float im = Iim[row * OSN + col];

  // apodization: a = sqrt(max(beta^2 - (pi*4*(i-160)/400)^2, 1e-12)); apod = a/sinh(a)
  float ay_arg = (float)(y - 160) * 0.031415926535897934f;   // pi/100
  float ax_arg = (float)(x - 160) * 0.031415926535897934f;
  float b2 = kBeta * kBeta;
  float ay = sqrtf(fmaxf(b2 - ay_arg * ay_arg, 1e-12f));
  float ax = sqrtf(fmaxf(b2 - ax_arg * ax_arg, 1e-12f));
  float apod = (ay / sinhf(ay)) * (ax / sinhf(ax));

  // scale: (osH*osW/sqrt(H*W)) * (1/(osH*osW)) = 1/320   (1/N of ifft not in W tables)
  float s = 0.003125f * apod;

  // out layout: (B, 1, 2, 320, 320)
  out[(((size_t)b * 2 + 0) * RESN + y) * RESN + x] = re * s;
  out[(((size_t)b * 2 + 1) * RESN + y) * RESN + x] = im * s;
}

// ---------------- host-side launcher ----------------
extern "C" void kernel_launch(void* const* d_in, const int* in_sizes, int n_in,
                              void* d_out, int out_size, void* d_ws, size_t ws_size,
                              hipStream_t stream) {
  const float* ksp  = (const float*)d_in[0];   // (16,1,320,320,2) f32
  const float* traj = (const float*)d_in[1];   // (25600,2) f32
  float* out = (float*)d_out;                  // (16,1,2,320,320) f32
  float* ws  = (float*)d_ws;

  // workspace layout (floats):
  float* Wre  = ws;                            //   160000
  float* Wim  = ws + 160000;                   //   160000
  float* grid = ws + 320000;                   // 5 120 000  (B*2*400*400) -- also stage-2 out
  float* T    = ws + 5440000;                  // 5 120 000
  // total: 10 560 000 floats = 42.24 MB

  hipMemsetAsync(grid, 0, (size_t)BATCH * 2 * NG2 * sizeof(float), stream);

  k_twiddle       <<<625,  256, 0, stream>>>(Wre, Wim);
  k_sample_scatter<<<1600, 256, 0, stream>>>(ksp, traj, grid);
  k_dft_stage1    <<<1250, 256, 0, stream>>>(grid, Wre, Wim, T);
  k_dft_stage2    <<<1250, 256, 0, stream>>>(T, Wre, Wim, grid);   // grid is dead -> reuse
  k_finalize      <<<6400, 256, 0, stream>>>(grid, out);
}